// hetero_actor_48232482734726
// MI455X (gfx1250) — compile-verified
//
#include <hip/hip_runtime.h>
#include <hip/hip_bf16.h>
#include <math.h>

// ---------------------------------------------------------------------------
// Hetero GNN actor head, fully algebraically fused for MI455X (gfx1250).
//
//   out = agg_tj@(Wrel_tj@Wo) + agg_jj@(Wrel_jj@Wo)
//       + xj@((Wroot_tj+Wroot_jj)@Wo) + ((brel_tj+brel_jj)@Wo + bo)
//   and segment_sum(x[src])@M == segment_sum((x@M)[src])
//
// => per-edge payload is 2 floats, the 64-dim hidden never materializes.
// Traffic: ~144MB edge indices (streamed, NT hint) + L2-resident 8MB/1.6MB/8MB
// node tables + 36M hw f32 atomics. ~7-10us at 23.3 TB/s.
//
// WMMA: torso transform (Nt x 11 @ 11x2) runs on V_WMMA_F32_16X16X4_F32,
// one wave per 16-row tile, K padded to 12 (3 chained WMMAs), f32 end-to-end.
// All WMMA operand loads are unconditionally in-bounds (clamped addresses);
// zero-padding is applied with value selects (v_cndmask), not EXEC branches.
// ---------------------------------------------------------------------------

typedef __attribute__((ext_vector_type(2))) float v2f;
typedef __attribute__((ext_vector_type(8))) float v8f;
typedef __attribute__((ext_vector_type(4))) int   v4i;

#define SP_BIAS  0.5413248546129181f   // log(expm1(1.0))
#define SCALE_LB 1.0e-4f

// ws const block layout (floats):
//  [0..21]  Mt   (11x2, i*2+o)   torso fused matrix  = Wt @ (Wrel_tj@Wo)
//  [22..23] ct                    torso fused bias    = bt @ (Wrel_tj@Wo)
//  [24..31] Mj   (2x4, i*4+c)    joint fused: cols 0-1 -> y (Wj@B), 2-3 -> z (Wj@C)
//  [32..35] cj   (y0,y1,z0,z1)
//  [36..37] dvec                  (brel_tj+brel_jj)@Wo + bo
//  [38..63] zero fill (so unconditional padded reads are well-defined)
#define WS_CONST_FLOATS 64

__device__ __forceinline__ void atomic_add_f32(float* p, float v) {
    // guaranteed hw f32 atomic (no CAS loop); s_endpgm implicitly waits idle
    asm volatile("global_atomic_add_f32 %0, %1, off" :: "v"(p), "v"(v) : "memory");
}

// ---- one-block fusion of all weight matrices --------------------------------
__global__ void k_fuse(const float* __restrict__ Wj,      const float* __restrict__ bj,
                       const float* __restrict__ Wt,      const float* __restrict__ bt,
                       const float* __restrict__ Wrel_tj, const float* __restrict__ brel_tj,
                       const float* __restrict__ Wroot_tj,
                       const float* __restrict__ Wrel_jj, const float* __restrict__ brel_jj,
                       const float* __restrict__ Wroot_jj,
                       const float* __restrict__ Wo,      const float* __restrict__ bo,
                       float* __restrict__ cst) {
    __shared__ float sA[22], sB[22], sC[22];   // A=Wrel_tj@Wo, B=Wrel_jj@Wo, C=(Wroot_tj+Wroot_jj)@Wo  (11x2)
    const int tid = threadIdx.x;
    if (tid < 22) {
        const int i = tid >> 1, o = tid & 1;
        float a = 0.f, b = 0.f, c = 0.f;
        for (int k = 0; k < 64; ++k) {
            const float w = Wo[k * 2 + o];
            a += Wrel_tj[i * 64 + k] * w;
            b += Wrel_jj[i * 64 + k] * w;
            c += (Wroot_tj[i * 64 + k] + Wroot_jj[i * 64 + k]) * w;
        }
        sA[tid] = a; sB[tid] = b; sC[tid] = c;
    }
    __syncthreads();
    if (tid < 22) {                                   // Mt = Wt @ A
        const int i = tid >> 1, o = tid & 1;
        float v = 0.f;
        for (int m = 0; m < 11; ++m) v += Wt[i * 11 + m] * sA[m * 2 + o];
        cst[tid] = v;
    } else if (tid < 24) {                            // ct = bt @ A
        const int o = tid - 22;
        float v = 0.f;
        for (int m = 0; m < 11; ++m) v += bt[m] * sA[m * 2 + o];
        cst[22 + o] = v;
    } else if (tid < 32) {                            // Mj: Wj@B | Wj@C
        const int e = tid - 24, i = e >> 2, c = e & 3;
        const float* S = (c < 2) ? sB : sC;
        const int o = c & 1;
        float v = 0.f;
        for (int m = 0; m < 11; ++m) v += Wj[i * 11 + m] * S[m * 2 + o];
        cst[24 + e] = v;
    } else if (tid < 36) {                            // cj: bj@B | bj@C
        const int c = tid - 32;
        const float* S = (c < 2) ? sB : sC;
        const int o = c & 1;
        float v = 0.f;
        for (int m = 0; m < 11; ++m) v += bj[m] * S[m * 2 + o];
        cst[32 + c] = v;
    } else if (tid < 38) {                            // dvec
        const int o = tid - 36;
        float v = bo[o];
        for (int k = 0; k < 64; ++k) v += (brel_tj[k] + brel_jj[k]) * Wo[k * 2 + o];
        cst[36 + o] = v;
    } else if (tid < WS_CONST_FLOATS) {               // zero pad region
        cst[tid] = 0.f;
    }
}

// ---- torso transform: yt = x_torso @ Mt + ct  via V_WMMA_F32_16X16X4_F32 ----
// One wave per 16-row tile. A: 16x12 (K padded from 11), B: 12x16 (cols 0,1 live).
// A layout (16x4 f32): v0 = K=k0   (lanes 0-15) / K=k0+2 (lanes 16-31)
//                      v1 = K=k0+1 (lanes 0-15) / K=k0+3 (lanes 16-31)
// B layout (4x16 f32): mirrored, N = lane&15.
// D layout: vgpr v -> M = v (lanes 0-15) / v+8 (lanes 16-31), N = lane&15.
__global__ void k_torso_wmma(const float* __restrict__ xt, const float* __restrict__ cst,
                             float* __restrict__ yt, int Nt, int numTiles) {
    const int tile = blockIdx.x * (blockDim.x >> 5) + (threadIdx.x >> 5);
    if (tile >= numTiles) return;                 // wave-uniform exit: EXEC stays full for WMMA
    const int lane = threadIdx.x & 31;
    const int half = lane >> 4;                   // 0: lanes 0-15, 1: lanes 16-31
    const int r    = lane & 15;                   // row-in-tile (A) / N column (B,D)
    const int row  = tile * 16 + r;
    const bool rowOK = row < Nt;
    const int rowC  = rowOK ? row : (Nt - 1);     // clamped: loads always in-bounds
    const float* xrow = xt + (long)rowC * 11;
    const bool nOK = r < 2;                       // live output columns

    v8f acc = {};
#pragma unroll
    for (int s = 0; s < 3; ++s) {
        const int k0 = s * 4 + half * 2;          // this half-wave's K pair
        const int c0 = (k0     < 11) ? k0     : 10;   // clamped column indices
        const int c1 = (k0 + 1 < 11) ? k0 + 1 : 10;
        // Unconditional loads (addresses always valid), zero-pad via selects.
        const float a0 = xrow[c0];
        const float a1 = xrow[c1];
        const float b0 = cst[k0 * 2 + r];             // idx <= 35 < 64: always safe
        const float b1 = cst[(k0 + 1) * 2 + r];       // idx <= 37 < 64: always safe
        v2f a, b;
        a.x = (rowOK && k0     < 11) ? a0 : 0.0f;
        a.y = (rowOK && k0 + 1 < 11) ? a1 : 0.0f;
        b.x = (nOK   && k0     < 11) ? b0 : 0.0f;     // Mt[k][n]
        b.y = (nOK   && k0 + 1 < 11) ? b1 : 0.0f;
        acc = __builtin_amdgcn_wmma_f32_16x16x4_f32(false, a, false, b,
                                                    (short)0, acc, false, false);
    }
    if (nOK) {
        const float bias = cst[22 + r];           // ct[n]
#pragma unroll
        for (int v = 0; v < 8; ++v) {
            const int rr = tile * 16 + v + half * 8;
            if (rr < Nt) yt[(long)rr * 2 + r] = acc[v] + bias;
        }
    }
}

// ---- joint transform: yj (edge payload) + acc init (root term + bias) -------
__global__ void k_joint(const float* __restrict__ xj, const float* __restrict__ cst,
                        float* __restrict__ yj, float* __restrict__ accv, int Nj) {
    const int j = blockIdx.x * blockDim.x + threadIdx.x;
    if (j >= Nj) return;
    const v2f x = *(const v2f*)(xj + (long)j * 2);
    const float m00 = cst[24], m01 = cst[25], m02 = cst[26], m03 = cst[27];
    const float m10 = cst[28], m11 = cst[29], m12 = cst[30], m13 = cst[31];
    v2f y, z;
    y.x = x.x * m00 + x.y * m10 + cst[32];
    y.y = x.x * m01 + x.y * m11 + cst[33];
    z.x = x.x * m02 + x.y * m12 + cst[34] + cst[36];
    z.y = x.x * m03 + x.y * m13 + cst[35] + cst[37];
    *(v2f*)(yj   + (long)j * 2) = y;
    *(v2f*)(accv + (long)j * 2) = z;
}

// ---- edge scatter-add: acc[dst] += yfeat[src]  (4 edges/thread, NT index stream)
__global__ void k_edges(const int* __restrict__ src, const int* __restrict__ dst,
                        const float* __restrict__ yfeat, float* __restrict__ accv, long E) {
    const long e0 = ((long)blockIdx.x * blockDim.x + threadIdx.x) * 4;
    if (e0 >= E) return;
    if (e0 + 4 <= E) {
        const v4i s = __builtin_nontemporal_load((const v4i*)(src + e0));
        const v4i d = __builtin_nontemporal_load((const v4i*)(dst + e0));
#pragma unroll
        for (int k = 0; k < 4; ++k) {
            const v2f y = *(const v2f*)(yfeat + (long)s[k] * 2);
            float* p = accv + (long)d[k] * 2;
            atomic_add_f32(p,     y.x);
            atomic_add_f32(p + 1, y.y);
        }
    } else {
        for (long e = e0; e < E; ++e) {
            const v2f y = *(const v2f*)(yfeat + (long)src[e] * 2);
            float* p = accv + (long)dst[e] * 2;
            atomic_add_f32(p,     y.x);
            atomic_add_f32(p + 1, y.y);
        }
    }
}

// ---- epilogue: loc / clipped softplus scale ---------------------------------
__global__ void k_final(const float* __restrict__ accv, float* __restrict__ out, int Nj) {
    const int j = blockIdx.x * blockDim.x + threadIdx.x;
    if (j >= Nj) return;
    const v2f a = *(const v2f*)(accv + (long)j * 2);
    out[j] = a.x;
    const float t  = a.y + SP_BIAS;
    const float sp = (t > 20.0f) ? t : log1pf(expf(t));
    out[(long)Nj + j] = fmaxf(sp, SCALE_LB);
}

extern "C" void kernel_launch(void* const* d_in, const int* in_sizes, int n_in,
                              void* d_out, int out_size, void* d_ws, size_t ws_size,
                              hipStream_t stream) {
    const float* x_joint  = (const float*)d_in[0];
    const float* x_torso  = (const float*)d_in[1];
    const int*   etj_src  = (const int*)  d_in[2];
    const int*   etj_dst  = (const int*)  d_in[3];
    const int*   ejj_src  = (const int*)  d_in[4];
    const int*   ejj_dst  = (const int*)  d_in[5];
    const float* Wj       = (const float*)d_in[6];
    const float* bj       = (const float*)d_in[7];
    const float* Wt       = (const float*)d_in[8];
    const float* bt       = (const float*)d_in[9];
    const float* Wrel_tj  = (const float*)d_in[10];
    const float* brel_tj  = (const float*)d_in[11];
    const float* Wroot_tj = (const float*)d_in[12];
    const float* Wrel_jj  = (const float*)d_in[13];
    const float* brel_jj  = (const float*)d_in[14];
    const float* Wroot_jj = (const float*)d_in[15];
    const float* Wo       = (const float*)d_in[16];
    const float* bo       = (const float*)d_in[17];

    const long Nj  = in_sizes[0] / 2;
    const long Nt  = in_sizes[1] / 11;
    const long Etj = in_sizes[2];
    const long Ejj = in_sizes[4];

    float* ws   = (float*)d_ws;
    float* cst  = ws;                              // fused weights
    float* yt   = ws + WS_CONST_FLOATS;            // Nt*2
    float* yj   = yt + 2 * Nt;                     // Nj*2
    float* accv = yj + 2 * Nj;                     // Nj*2
    float* out  = (float*)d_out;

    k_fuse<<<1, 64, 0, stream>>>(Wj, bj, Wt, bt, Wrel_tj, brel_tj, Wroot_tj,
                                 Wrel_jj, brel_jj, Wroot_jj, Wo, bo, cst);

    const int tiles = (int)((Nt + 15) / 16);
    k_torso_wmma<<<(tiles + 7) / 8, 256, 0, stream>>>(x_torso, cst, yt, (int)Nt, tiles);

    k_joint<<<(unsigned)((Nj + 255) / 256), 256, 0, stream>>>(x_joint, cst, yj, accv, (int)Nj);

    k_edges<<<(unsigned)(((Etj + 3) / 4 + 255) / 256), 256, 0, stream>>>(etj_src, etj_dst, yt, accv, Etj);
    k_edges<<<(unsigned)(((Ejj + 3) / 4 + 255) / 256), 256, 0, stream>>>(ejj_src, ejj_dst, yj, accv, Ejj);

    k_final<<<(unsigned)((Nj + 255) / 256), 256, 0, stream>>>(accv, out, (int)Nj);
}